// NerfRenderer_23888608100544
// MI455X (gfx1250) — compile-verified
//
#include <hip/hip_runtime.h>
#include <stdint.h>

// NeRF weights + top-k sample selection for MI455X (gfx1250).
// One wave32 per ray:
//   - async global->LDS staging of the 3KB sigma row (ASYNCcnt, TH_LOAD_NT)
//   - shuffle prefix-sum for transmittance, 24x v_exp_f32 per lane
//   - ballot + scalar-pipe bitwise k-th-value selection (top 16 float bits)
//   - selected z values compacted into LDS, streamed out with
//     global_store_async_from_lds_b128 (coalesced 512B async NT stores)
// All global traffic is touch-once streaming (480 MiB total), so every access
// carries a non-temporal hint to keep the 192MB L2 unpolluted.

#define N_SAMPLES   768
#define PER_LANE    24          // 768 / 32
#define N_FREE_LANE 18          // per lane: 24 cols, 6 forced (col%4==0), 18 free
#define K_SEL       192         // INNER_STEPS/2 top-k among free columns
#define INNER_STEPS 384
#define WAVES_PER_BLOCK 8
#define THREADS_PER_BLOCK (WAVES_PER_BLOCK * 32)

// log10(0.05) and the (constant) linspace step (hi-lo)/n, which also equals delta.
#define Z_LO   (-1.3010299956639813f)
#define Z_STEP (1.3010299956639813f / 768.0f)

typedef __attribute__((ext_vector_type(4))) float v4f;   // native vector: OK for NT builtins

__device__ __forceinline__ int wave_incl_scan_i32(int v, int lane) {
#pragma unroll
    for (int d = 1; d < 32; d <<= 1) {
        int t = __shfl_up(v, d, 32);
        if (lane >= d) v += t;
    }
    return v;
}

__device__ __forceinline__ float wave_incl_scan_f32(float v, int lane) {
#pragma unroll
    for (int d = 1; d < 32; d <<= 1) {
        float t = __shfl_up(v, d, 32);
        if (lane >= d) v += t;
    }
    return v;
}

__global__ __launch_bounds__(THREADS_PER_BLOCK)
void nerf_weights_select_kernel(const float* __restrict__ sigmas,
                                float* __restrict__ z_out,   // n_rays x 384
                                float* __restrict__ w_out,   // n_rays x 768
                                int n_rays)
{
    __shared__ __align__(16) float smem[WAVES_PER_BLOCK * N_SAMPLES]; // 24 KB

    const int lane   = (int)(threadIdx.x & 31u);
    const int waveId = (int)(threadIdx.x >> 5u);
    const int ray    = (int)blockIdx.x * WAVES_PER_BLOCK + waveId;
    if (ray >= n_rays) return;                       // wave-uniform exit

    const float* rowp = sigmas + (size_t)ray * N_SAMPLES;
    float* srow = &smem[waveId * N_SAMPLES];
    // Flat shared pointer = {aperture_hi, lds_byte_offset}: low 32 bits are the
    // workgroup-relative LDS address the async VDST/VSRC operands want.
    const unsigned lds_row = (unsigned)(uintptr_t)srow;

    // ---- CDNA5 async global->LDS copy of this wave's 3 KB sigma row (6 x 512B).
    // Touch-once stream -> non-temporal load hint.
#pragma unroll
    for (int j = 0; j < 6; ++j) {
        unsigned voff = (unsigned)(lane * 16 + j * 512);
        unsigned lptr = lds_row + voff;
        asm volatile("global_load_async_to_lds_b128 %0, %1, %2 th:TH_LOAD_NT"
                     :
                     : "v"(lptr), "v"(voff), "s"(rowp)
                     : "memory");
    }
    asm volatile("s_wait_asynccnt 0x0" ::: "memory");

    // Each lane reads its 24 contiguous sigmas from LDS (6 x ds_load_b128).
    float sig[PER_LANE];
    {
        const v4f* sp = (const v4f*)(srow + lane * PER_LANE);
#pragma unroll
        for (int j = 0; j < 6; ++j) {
            v4f v = sp[j];
            sig[4*j+0] = v.x; sig[4*j+1] = v.y; sig[4*j+2] = v.z; sig[4*j+3] = v.w;
        }
    }

    // Wave exclusive prefix sum of per-lane sigma sums -> transmittance seed.
    float local_sum = 0.f;
#pragma unroll
    for (int i = 0; i < PER_LANE; ++i) local_sum += sig[i];
    const float incl = wave_incl_scan_f32(local_sum, lane);
    const float excl = incl - local_sum;

    // w_i = T_i - T_{i+1}, T_i = exp(-delta * prefix_sigma_i); w[767] = 0.
    float T = __expf(-Z_STEP * excl);
    float w[PER_LANE];
#pragma unroll
    for (int i = 0; i < PER_LANE; ++i) {
        float Tn = T * __expf(-Z_STEP * sig[i]);
        w[i] = T - Tn;
        T = Tn;
    }
    if (lane == 31) w[PER_LANE - 1] = 0.0f;          // delta[last] == 0 -> alpha 0

    // Store the weights row: coalesced 16B non-temporal stores from registers.
    {
        v4f* wp = (v4f*)(w_out + (size_t)ray * N_SAMPLES + lane * PER_LANE);
#pragma unroll
        for (int j = 0; j < 6; ++j) {
            v4f v = { w[4*j+0], w[4*j+1], w[4*j+2], w[4*j+3] };
            __builtin_nontemporal_store(v, &wp[j]);
        }
    }

    // ---- k-th largest free weight via bitwise search on the top 16 float bits.
    // Weights are in [0,1] so the uint bit pattern is order-preserving and the
    // truncated key fits in 14 bits. Counting uses wave32 ballot: the popcount
    // accumulation runs on the scalar pipe and co-issues with VALU.
    // Free columns are i%4!=0 within this lane's 24 (lane start col % 4 == 0).
    unsigned key[N_FREE_LANE];
    {
        int nf = 0;
#pragma unroll
        for (int i = 0; i < PER_LANE; ++i)
            if ((i & 3) != 0) key[nf++] = __float_as_uint(w[i]) >> 16;
    }

    unsigned T16 = 0u;
    for (int b = 13; b >= 0; --b) {
        const unsigned cand = T16 | (1u << b);
        int cnt = 0;
#pragma unroll
        for (int t = 0; t < N_FREE_LANE; ++t) {
            unsigned long long m = __ballot(key[t] >= cand);
            cnt += __popcll(m);
        }
        if (cnt >= K_SEL) T16 = cand;                // max T16 with count_ge >= k
    }

    // Per-lane strictly-greater / equal counts, wave scans for tie-break ranks.
    int gt_l = 0, eq_l = 0;
#pragma unroll
    for (int t = 0; t < N_FREE_LANE; ++t) {
        gt_l += (key[t] >  T16);
        eq_l += (key[t] == T16);
    }
    const int packed   = (gt_l << 16) | eq_l;        // counts <= 576, fit 16 bits
    const int pincl    = wave_incl_scan_i32(packed, lane);
    const int eq_excl  = (pincl & 0xFFFF) - eq_l;
    const int gt_total = __shfl(pincl, 31, 32) >> 16;
    const int fill     = K_SEL - gt_total;           // > 0 and <= total equals

    int eq_sel_l = fill - eq_excl;
    if (eq_sel_l < 0)    eq_sel_l = 0;
    if (eq_sel_l > eq_l) eq_sel_l = eq_l;
    const int sel_l    = 6 + gt_l + eq_sel_l;        // 6 forced + chosen free
    const int sel_incl = wave_incl_scan_i32(sel_l, lane);
    int out_off = sel_incl - sel_l;                  // row offset, ascending-col order

    // ---- Compact selected z values into LDS (sigma staging region is dead now;
    // same-wave DS ops are ordered, so no wait is needed before overwriting).
    int eq_rank = eq_excl;
    int nf = 0;
#pragma unroll
    for (int i = 0; i < PER_LANE; ++i) {
        const int col = lane * PER_LANE + i;
        bool sel;
        if ((i & 3) == 0) {
            sel = true;
        } else {
            const unsigned kk = key[nf++];
            if (kk > T16)       sel = true;
            else if (kk == T16) { sel = (eq_rank < fill); ++eq_rank; }
            else                sel = false;
        }
        if (sel) {
            srow[out_off] = Z_LO + Z_STEP * (float)col;   // ds_store_b32, compacted
            ++out_off;
        }
    }

    // ---- Stream the compacted 1.5 KB z row out with async LDS->global stores.
    // DS writes and ASYNC ops use different counters -> drain DScnt first.
    float* zrow = z_out + (size_t)ray * INNER_STEPS;
    asm volatile("s_wait_dscnt 0x0" ::: "memory");
#pragma unroll
    for (int j = 0; j < 3; ++j) {
        unsigned voff = (unsigned)(lane * 16 + j * 512);
        unsigned lptr = lds_row + voff;
        asm volatile("global_store_async_from_lds_b128 %0, %1, %2 th:TH_STORE_NT"
                     :
                     : "v"(voff), "v"(lptr), "s"(zrow)
                     : "memory");
    }
    // Make sure the DMA has consumed this wave's LDS before it terminates
    // (S_ENDPGM's implicit wait-idle also covers this; explicit is free).
    asm volatile("s_wait_asynccnt 0x0" ::: "memory");
}

extern "C" void kernel_launch(void* const* d_in, const int* in_sizes, int n_in,
                              void* d_out, int out_size, void* d_ws, size_t ws_size,
                              hipStream_t stream) {
    (void)n_in; (void)d_ws; (void)ws_size; (void)out_size;
    const float* sigmas = (const float*)d_in[0];
    const int n_rays = in_sizes[0] / N_SAMPLES;

    // Output tuple order: z_vals_log_inner (n_rays x 384) then weights (n_rays x 768).
    float* z_out = (float*)d_out;
    float* w_out = (float*)d_out + (size_t)n_rays * INNER_STEPS;

    const int blocks = (n_rays + WAVES_PER_BLOCK - 1) / WAVES_PER_BLOCK;
    nerf_weights_select_kernel<<<blocks, THREADS_PER_BLOCK, 0, stream>>>(
        sigmas, z_out, w_out, n_rays);
}